// CombinedModel_25598005084994
// MI455X (gfx1250) — compile-verified
//
#include <hip/hip_runtime.h>
#include <math.h>

#define TEXT_DIM 5000
#define GIN 128

typedef __attribute__((ext_vector_type(2))) float v2f;
typedef __attribute__((ext_vector_type(8))) float v8f;

__device__ __forceinline__ v8f wmma_f32_4(v2f a, v2f b, v8f c) {
    return __builtin_amdgcn_wmma_f32_16x16x4_f32(false, a, false, b, (short)0, c,
                                                 false, false);
}

// ---------------------------------------------------------------------------
// Encoder: h1 = relu(text_x @ W_e1 + b_e1)  (4096x5000 @ 5000x8)
// fused:   z_text = relu(h1 @ W_e2 + b_e2)       -> zbuf[:, 0:4]
//          hdec   = relu(z_text @ W_d1 + b_d1)   -> hdec
// One block = 8 waves; all 8 waves cooperate on one 16-row tile, splitting K.
// B-operand is built branch-free: unconditional load from (n&7) column, then
// multiply by a 0/1 lane mask -> no exec-mask branches in the WMMA loop.
// ---------------------------------------------------------------------------
__global__ __launch_bounds__(256) void k_encoder(
    const float* __restrict__ text_x,
    const float* __restrict__ W_e1, const float* __restrict__ b_e1,
    const float* __restrict__ W_e2, const float* __restrict__ b_e2,
    const float* __restrict__ W_d1, const float* __restrict__ b_d1,
    float* __restrict__ zbuf, float* __restrict__ hdec)
{
    __shared__ float red[8][16][8];   // per-wave partial 16x8 tiles
    __shared__ float h1s[16][8];

    const int w    = threadIdx.x >> 5;
    const int lane = threadIdx.x & 31;
    const int n    = lane & 15;
    const int hi   = lane >> 4;
    const int m0   = blockIdx.x * 16;
    const float bmask = (n < 8) ? 1.0f : 0.0f;

    const float* arow = text_x + (size_t)(m0 + n) * TEXT_DIM;  // A row for this lane
    const float* bcol = W_e1 + (n & 7);                        // safe B column

    v8f acc = {};
    const int nsteps = TEXT_DIM / 4;         // 1250 K-steps of 4
    for (int s = w; s < nsteps; s += 8) {
        const int k = s * 4 + 2 * hi;
        v2f a;
        a.x = arow[k];
        a.y = arow[k + 1];
        __builtin_prefetch(&arow[k + 32], 0, 1);
        v2f b;
        b.x = bcol[k * 8] * bmask;
        b.y = bcol[(k + 1) * 8] * bmask;
        acc = wmma_f32_4(a, b, acc);
    }

    if (n < 8) {
#pragma unroll
        for (int r = 0; r < 8; ++r) red[w][r + 8 * hi][n] = acc[r];
    }
    __syncthreads();

    // cross-wave reduction + bias + relu
    if (threadIdx.x < 128) {
        const int row = threadIdx.x >> 3, col = threadIdx.x & 7;
        float s = 0.0f;
#pragma unroll
        for (int ww = 0; ww < 8; ++ww) s += red[ww][row][col];
        s += b_e1[col];
        h1s[row][col] = s > 0.0f ? s : 0.0f;
    }
    __syncthreads();

    // tiny fused MLPs: 8 -> 4 (z_text) -> 8 (hdec)
    if (threadIdx.x < 16) {
        const int row = threadIdx.x;
        float zt[4];
#pragma unroll
        for (int c = 0; c < 4; ++c) {
            float s = b_e2[c];
#pragma unroll
            for (int j = 0; j < 8; ++j) s += h1s[row][j] * W_e2[j * 4 + c];
            zt[c] = s > 0.0f ? s : 0.0f;
            zbuf[(m0 + row) * 8 + c] = zt[c];
        }
#pragma unroll
        for (int j = 0; j < 8; ++j) {
            float s = b_d1[j];
#pragma unroll
            for (int c = 0; c < 4; ++c) s += zt[c] * W_d1[c * 8 + j];
            hdec[(m0 + row) * 8 + j] = s > 0.0f ? s : 0.0f;
        }
    }
}

// ---------------------------------------------------------------------------
// Decoder: x_hat = sigmoid(hdec @ W_d2 + b_d2)  (4096x8 @ 8x5000)
// One wave per 16x16 output tile; K=8 -> 2 WMMA steps. Tail tile handled with
// clamped addresses + mask multiply (branch-free through the WMMAs).
// ---------------------------------------------------------------------------
__global__ __launch_bounds__(256) void k_decoder(
    const float* __restrict__ hdec,
    const float* __restrict__ W_d2, const float* __restrict__ b_d2,
    float* __restrict__ x_hat, int Mrows)
{
    const int w    = threadIdx.x >> 5;
    const int lane = threadIdx.x & 31;
    const int n    = lane & 15;
    const int hi   = lane >> 4;
    const int NT   = (TEXT_DIM + 15) / 16;            // 313
    const int tile = blockIdx.x * 8 + w;
    if (tile >= (Mrows / 16) * NT) return;            // wave-uniform exit
    const int m0 = (tile / NT) * 16;
    const int n0 = (tile % NT) * 16;
    const int col  = n0 + n;
    const bool ok  = col < TEXT_DIM;
    const int colc = ok ? col : (TEXT_DIM - 1);       // clamped, always valid
    const float cmask = ok ? 1.0f : 0.0f;

    const float* arow = hdec + (size_t)(m0 + n) * 8;
    const float* bcol = W_d2 + colc;

    v8f acc = {};
#pragma unroll
    for (int k0 = 0; k0 < 8; k0 += 4) {
        const int k = k0 + 2 * hi;
        v2f a;
        a.x = arow[k];
        a.y = arow[k + 1];
        v2f b;
        b.x = bcol[k * TEXT_DIM] * cmask;
        b.y = bcol[(k + 1) * TEXT_DIM] * cmask;
        acc = wmma_f32_4(a, b, acc);
    }
    if (ok) {
        const float bb = b_d2[col];
#pragma unroll
        for (int r = 0; r < 8; ++r) {
            const int row = m0 + r + 8 * hi;
            const float v = acc[r] + bb;
            x_hat[row * TEXT_DIM + col] = 1.0f / (1.0f + __expf(-v));
        }
    }
}

// ---------------------------------------------------------------------------
// GNN linear 1: g1lin = node_x @ W_g1  (200000x128 @ 128x8), WMMA, K=128.
// ---------------------------------------------------------------------------
__global__ __launch_bounds__(256) void k_gnn_lin1(
    const float* __restrict__ node_x, const float* __restrict__ W_g1,
    float* __restrict__ g1lin, int Ntiles, int Nn)
{
    const int w    = threadIdx.x >> 5;
    const int lane = threadIdx.x & 31;
    const int n    = lane & 15;
    const int hi   = lane >> 4;
    const int tile = blockIdx.x * 8 + w;
    if (tile >= Ntiles) return;                       // wave-uniform exit
    const int m0 = tile * 16;
    int rowA = m0 + n;
    if (rowA >= Nn) rowA = Nn - 1;                    // clamp (N % 16 == 0 here)
    const float bmask = (n < 8) ? 1.0f : 0.0f;

    const float* arow = node_x + (size_t)rowA * GIN;
    const float* bcol = W_g1 + (n & 7);

    v8f acc = {};
#pragma unroll 4
    for (int k0 = 0; k0 < GIN; k0 += 4) {
        const int k = k0 + 2 * hi;
        v2f a;
        a.x = arow[k];
        a.y = arow[k + 1];
        v2f b;
        b.x = bcol[k * 8] * bmask;
        b.y = bcol[(k + 1) * 8] * bmask;
        acc = wmma_f32_4(a, b, acc);
    }
    if (n < 8) {
#pragma unroll
        for (int r = 0; r < 8; ++r) {
            const int row = m0 + r + 8 * hi;
            if (row < Nn) g1lin[(size_t)row * 8 + n] = acc[r];
        }
    }
}

// ---------------------------------------------------------------------------
// GCN plumbing
// ---------------------------------------------------------------------------
__global__ void k_deg_init(float* __restrict__ deg, int Nn) {
    int i = blockIdx.x * blockDim.x + threadIdx.x;
    if (i < Nn) deg[i] = 1.0f;                        // self-loop
}

__global__ void k_deg_scatter(const int* __restrict__ dst, float* __restrict__ deg, int En) {
    int e = blockIdx.x * blockDim.x + threadIdx.x;
    if (e < En) atomicAdd(&deg[dst[e]], 1.0f);
}

__global__ void k_dinv(float* __restrict__ deg, int Nn) {
    int i = blockIdx.x * blockDim.x + threadIdx.x;
    if (i < Nn) deg[i] = rsqrtf(deg[i]);              // deg >= 1 always
}

// self-loop contribution initializes the accumulator (assignment, not atomic)
__global__ void k_selfinit1(const float* __restrict__ g1lin, const float* __restrict__ dinv,
                            float* __restrict__ h1agg, int Nn) {
    int i = blockIdx.x * blockDim.x + threadIdx.x;
    if (i >= Nn) return;
    const float di = dinv[i];
    const float di2 = di * di;
    const float4* gs = (const float4*)(g1lin + (size_t)i * 8);
    float4* hd = (float4*)(h1agg + (size_t)i * 8);
    float4 a = gs[0], b = gs[1];
    a.x *= di2; a.y *= di2; a.z *= di2; a.w *= di2;
    b.x *= di2; b.y *= di2; b.z *= di2; b.w *= di2;
    hd[0] = a; hd[1] = b;
}

__global__ void k_scatter1(const int* __restrict__ src, const int* __restrict__ dst,
                           const float* __restrict__ dinv, const float* __restrict__ g1lin,
                           float* __restrict__ h1agg, int En) {
    int e = blockIdx.x * blockDim.x + threadIdx.x;
    if (e >= En) return;
    const int s = src[e], d = dst[e];
    const float wgt = dinv[s] * dinv[d];
    const float4* gs = (const float4*)(g1lin + (size_t)s * 8);
    const float4 a = gs[0], b = gs[1];
    float* hd = h1agg + (size_t)d * 8;
    atomicAdd(&hd[0], a.x * wgt);
    atomicAdd(&hd[1], a.y * wgt);
    atomicAdd(&hd[2], a.z * wgt);
    atomicAdd(&hd[3], a.w * wgt);
    atomicAdd(&hd[4], b.x * wgt);
    atomicAdd(&hd[5], b.y * wgt);
    atomicAdd(&hd[6], b.z * wgt);
    atomicAdd(&hd[7], b.w * wgt);
}

// h1 = relu(h1agg + b_g1); h2lin = h1 @ W_g2; h2agg initialized with self-loop term
__global__ void k_h2lin(const float* __restrict__ h1agg, const float* __restrict__ b_g1,
                        const float* __restrict__ W_g2, const float* __restrict__ dinv,
                        float* __restrict__ h2lin, float* __restrict__ h2agg, int Nn) {
    int i = blockIdx.x * blockDim.x + threadIdx.x;
    if (i >= Nn) return;
    const float4* ha = (const float4*)(h1agg + (size_t)i * 8);
    const float4 a = ha[0], b = ha[1];
    float h[8] = {a.x, a.y, a.z, a.w, b.x, b.y, b.z, b.w};
#pragma unroll
    for (int j = 0; j < 8; ++j) {
        const float v = h[j] + b_g1[j];
        h[j] = v > 0.0f ? v : 0.0f;
    }
    const float di = dinv[i];
    const float di2 = di * di;
    float4 lin, agg;
    float* linp = (float*)&lin;
    float* aggp = (float*)&agg;
#pragma unroll
    for (int c = 0; c < 4; ++c) {
        float s = 0.0f;
#pragma unroll
        for (int j = 0; j < 8; ++j) s += h[j] * W_g2[j * 4 + c];
        linp[c] = s;
        aggp[c] = s * di2;
    }
    ((float4*)(h2lin + (size_t)i * 4))[0] = lin;
    ((float4*)(h2agg + (size_t)i * 4))[0] = agg;
}

__global__ void k_scatter2(const int* __restrict__ src, const int* __restrict__ dst,
                           const float* __restrict__ dinv, const float* __restrict__ h2lin,
                           float* __restrict__ h2agg, int En) {
    int e = blockIdx.x * blockDim.x + threadIdx.x;
    if (e >= En) return;
    const int s = src[e], d = dst[e];
    const float wgt = dinv[s] * dinv[d];
    const float4 hs = ((const float4*)(h2lin + (size_t)s * 4))[0];
    float* hd = h2agg + (size_t)d * 4;
    atomicAdd(&hd[0], hs.x * wgt);
    atomicAdd(&hd[1], hs.y * wgt);
    atomicAdd(&hd[2], hs.z * wgt);
    atomicAdd(&hd[3], hs.w * wgt);
}

__global__ void k_poolzero(float* __restrict__ sums, float* __restrict__ cnt, int Bn) {
    int i = blockIdx.x * blockDim.x + threadIdx.x;
    if (i >= Bn) return;
    ((float4*)(sums + i * 4))[0] = make_float4(0.f, 0.f, 0.f, 0.f);
    cnt[i] = 0.0f;
}

__global__ void k_pool(const float* __restrict__ h2agg, const float* __restrict__ b_g2,
                       const int* __restrict__ batch, float* __restrict__ sums,
                       float* __restrict__ cnt, int Nn) {
    int i = blockIdx.x * blockDim.x + threadIdx.x;
    if (i >= Nn) return;
    const int b = batch[i];
    const float4 v = ((const float4*)(h2agg + (size_t)i * 4))[0];
    atomicAdd(&sums[b * 4 + 0], v.x + b_g2[0]);
    atomicAdd(&sums[b * 4 + 1], v.y + b_g2[1]);
    atomicAdd(&sums[b * 4 + 2], v.z + b_g2[2]);
    atomicAdd(&sums[b * 4 + 3], v.w + b_g2[3]);
    atomicAdd(&cnt[b], 1.0f);
}

__global__ void k_final(const float* __restrict__ sums, const float* __restrict__ cnt,
                        const float* __restrict__ W_c, const float* __restrict__ b_c,
                        float* __restrict__ zbuf, float* __restrict__ logits, int Bn) {
    int i = blockIdx.x * blockDim.x + threadIdx.x;
    if (i >= Bn) return;
    float c = cnt[i];
    c = c > 1.0f ? c : 1.0f;
    float z[8];
#pragma unroll
    for (int j = 0; j < 4; ++j) z[j] = zbuf[i * 8 + j];
#pragma unroll
    for (int j = 0; j < 4; ++j) {
        z[4 + j] = sums[i * 4 + j] / c;
        zbuf[i * 8 + 4 + j] = z[4 + j];
    }
    float lg = b_c[0];
#pragma unroll
    for (int j = 0; j < 8; ++j) lg += z[j] * W_c[j];
    logits[i] = lg;
}

// ---------------------------------------------------------------------------
extern "C" void kernel_launch(void* const* d_in, const int* in_sizes, int n_in,
                              void* d_out, int out_size, void* d_ws, size_t ws_size,
                              hipStream_t stream) {
    const float* text_x = (const float*)d_in[0];
    const float* node_x = (const float*)d_in[1];
    const int*   edge   = (const int*)d_in[2];
    const int*   batch  = (const int*)d_in[3];
    const float* W_e1 = (const float*)d_in[4];
    const float* b_e1 = (const float*)d_in[5];
    const float* W_e2 = (const float*)d_in[6];
    const float* b_e2 = (const float*)d_in[7];
    const float* W_d1 = (const float*)d_in[8];
    const float* b_d1 = (const float*)d_in[9];
    const float* W_d2 = (const float*)d_in[10];
    const float* b_d2 = (const float*)d_in[11];
    const float* W_g1 = (const float*)d_in[12];
    const float* b_g1 = (const float*)d_in[13];
    const float* W_g2 = (const float*)d_in[14];
    const float* b_g2 = (const float*)d_in[15];
    const float* W_c  = (const float*)d_in[16];
    const float* b_c  = (const float*)d_in[17];

    const int Bn = in_sizes[0] / TEXT_DIM;   // 4096
    const int Nn = in_sizes[1] / GIN;        // 200000
    const int En = in_sizes[2] / 2;          // 6,400,000

    const int* src = edge;
    const int* dst = edge + En;

    float* x_hat  = (float*)d_out;
    float* logits = x_hat + (size_t)Bn * TEXT_DIM;
    float* zbuf   = logits + Bn;

    float* ws    = (float*)d_ws;
    float* hdec  = ws;  ws += (size_t)Bn * 8;
    float* deg   = ws;  ws += Nn;            // reused in-place as dinv
    float* g1lin = ws;  ws += (size_t)Nn * 8;
    float* h1agg = ws;  ws += (size_t)Nn * 8;
    float* h2lin = ws;  ws += (size_t)Nn * 4;
    float* h2agg = ws;  ws += (size_t)Nn * 4;
    float* sums  = ws;  ws += (size_t)Bn * 4;
    float* cnt   = ws;

    const int T = 256;
    const int nblk = (Nn + T - 1) / T;
    const int eblk = (En + T - 1) / T;
    const int bblk = (Bn + T - 1) / T;

    // --- dense text path (WMMA) ---
    k_encoder<<<Bn / 16, 256, 0, stream>>>(text_x, W_e1, b_e1, W_e2, b_e2,
                                           W_d1, b_d1, zbuf, hdec);
    const int NT = (TEXT_DIM + 15) / 16;
    const int dtiles = (Bn / 16) * NT;
    k_decoder<<<(dtiles + 7) / 8, 256, 0, stream>>>(hdec, W_d2, b_d2, x_hat, Bn);

    // --- GNN path ---
    const int Ntiles = (Nn + 15) / 16;
    k_gnn_lin1<<<(Ntiles + 7) / 8, 256, 0, stream>>>(node_x, W_g1, g1lin, Ntiles, Nn);
    k_deg_init<<<nblk, T, 0, stream>>>(deg, Nn);
    k_deg_scatter<<<eblk, T, 0, stream>>>(dst, deg, En);
    k_dinv<<<nblk, T, 0, stream>>>(deg, Nn);
    k_selfinit1<<<nblk, T, 0, stream>>>(g1lin, deg, h1agg, Nn);
    k_scatter1<<<eblk, T, 0, stream>>>(src, dst, deg, g1lin, h1agg, En);
    k_h2lin<<<nblk, T, 0, stream>>>(h1agg, b_g1, W_g2, deg, h2lin, h2agg, Nn);
    k_scatter2<<<eblk, T, 0, stream>>>(src, dst, deg, h2lin, h2agg, En);
    k_poolzero<<<bblk, T, 0, stream>>>(sums, cnt, Bn);
    k_pool<<<nblk, T, 0, stream>>>(h2agg, b_g2, batch, sums, cnt, Nn);
    k_final<<<bblk, T, 0, stream>>>(sums, cnt, W_c, b_c, zbuf, logits, Bn);
}